// HashRouter_43490838839446
// MI455X (gfx1250) — compile-verified
//
#include <hip/hip_runtime.h>

typedef __attribute__((ext_vector_type(2))) float v2f;
typedef __attribute__((ext_vector_type(8))) float v8f;

#define NUM_TOKENS 131072
#define HIDDEN     2048
#define NUM_EXPERTS 64
#define PROJ_DIM    32

// ---------------------------------------------------------------------------
// Zero the logits+probs region (2 * 131072 * 64 floats = 64 MB) with float4
// stores. Runs before the router kernel on the same stream.
// ---------------------------------------------------------------------------
__global__ void HashRouter_zero_kernel(float4* __restrict__ p, int n4) {
    int i = blockIdx.x * blockDim.x + threadIdx.x;
    if (i < n4) p[i] = make_float4(0.f, 0.f, 0.f, 0.f);
}

// ---------------------------------------------------------------------------
// Projection (fp32 WMMA) + FNV hash routing.
// 256 threads = 8 waves; each wave computes a 16-token x 32-proj tile with
// V_WMMA_F32_16X16X4_F32, spills it to LDS, then 16 lanes hash one token each.
// ---------------------------------------------------------------------------
__global__ __launch_bounds__(256) void HashRouter_route_kernel(
    const float* __restrict__ x,
    const float* __restrict__ Wp,
    const float* __restrict__ bp,
    float* __restrict__ out)
{
    // LDS tile: 8 waves x 16 rows x 32 cols, padded to 33 (conflict-free
    // column-major reads during the hash phase). 16.5 KB of the 320 KB WGP LDS.
    __shared__ float tile[8][16][33];

    const int tid   = threadIdx.x;
    const int wave  = tid >> 5;        // 0..7
    const int lane  = tid & 31;
    const int l16   = lane & 15;
    const int khalf = lane >> 4;       // 0: K={0,1}, 1: K={2,3} within each 4-chunk
    const int m0    = (blockIdx.x * 8 + wave) * 16;   // first token of this wave

    // A fragment: row (m0+l16), columns kb + 4*s + 2*khalf + {0,1}
    const float* arow  = x  + (size_t)(m0 + l16) * HIDDEN + 2 * khalf;
    // B fragments: Wp row n, same K striping (B = Wp^T slice, K x N)
    const float* brow0 = Wp + (size_t)(l16)      * HIDDEN + 2 * khalf;
    const float* brow1 = Wp + (size_t)(16 + l16) * HIDDEN + 2 * khalf;

    v8f c0 = {0.f, 0.f, 0.f, 0.f, 0.f, 0.f, 0.f, 0.f};
    v8f c1 = {0.f, 0.f, 0.f, 0.f, 0.f, 0.f, 0.f, 0.f};

    for (int kb = 0; kb < HIDDEN; kb += 16) {
        // prefetch ~1KB ahead in this lane's x row (gfx1250 global_prefetch_b8)
        if (kb + 256 < HIDDEN) __builtin_prefetch(arow + kb + 256, 0, 1);
#pragma unroll
        for (int s = 0; s < 4; ++s) {
            v2f a  = *(const v2f*)(arow  + kb + 4 * s);
            v2f b0 = *(const v2f*)(brow0 + kb + 4 * s);
            v2f b1 = *(const v2f*)(brow1 + kb + 4 * s);
            // D = A(16x4) * B(4x16) + C, fp32 matrix pipe
            c0 = __builtin_amdgcn_wmma_f32_16x16x4_f32(
                     false, a, false, b0, (short)0, c0, false, false);
            c1 = __builtin_amdgcn_wmma_f32_16x16x4_f32(
                     false, a, false, b1, (short)0, c1, false, false);
        }
    }

    // Bias: accumulator layout has N = l16 (+16 for c1) constant per lane.
    const float bias0 = bp[l16];
    const float bias1 = bp[16 + l16];
    // C/D layout: VGPR r -> M = r + 8*khalf, N = l16.
    const int mbase = 8 * khalf;
#pragma unroll
    for (int r = 0; r < 8; ++r) {
        tile[wave][mbase + r][l16]      = c0[r] + bias0;
        tile[wave][mbase + r][16 + l16] = c1[r] + bias1;
    }

    __syncthreads();

    // Hash phase: 16 lanes per wave, one token each. Sequential FNV over the
    // 32 projected words (bit patterns), then mix in the position.
    if (lane < 16) {
        const int token = m0 + lane;
        unsigned h = 2166136261u;
#pragma unroll
        for (int j = 0; j < 32; ++j) {
            h = (h ^ __float_as_uint(tile[wave][lane][j])) * 2654435761u;
            h = h ^ (h >> 15);
        }
        h = (h ^ (unsigned)token) * 2246822519u;
        h = h ^ (h >> 13);
        const int expert = (int)(h & (NUM_EXPERTS - 1));

        // Output layout (flat, fp32): [logits | probs | expert_idx | expert_w]
        float* probs = out + (size_t)NUM_TOKENS * NUM_EXPERTS;
        float* eidx  = out + 2 * (size_t)NUM_TOKENS * NUM_EXPERTS;
        float* ewt   = eidx + NUM_TOKENS;

        probs[(size_t)token * NUM_EXPERTS + expert] = 1.0f;
        eidx[token] = (float)expert;
        ewt[token]  = 1.0f;
    }
}

extern "C" void kernel_launch(void* const* d_in, const int* in_sizes, int n_in,
                              void* d_out, int out_size, void* d_ws, size_t ws_size,
                              hipStream_t stream) {
    const float* x  = (const float*)d_in[0];   // (131072, 2048)
    const float* Wp = (const float*)d_in[1];   // (32, 2048)
    const float* bp = (const float*)d_in[2];   // (32,)
    float* out = (float*)d_out;

    // 1) zero logits + probs (16,777,216 floats = 4,194,304 float4)
    const int n4 = (NUM_TOKENS * NUM_EXPERTS * 2) / 4;
    HashRouter_zero_kernel<<<dim3((n4 + 255) / 256), dim3(256), 0, stream>>>(
        (float4*)out, n4);

    // 2) projection + hash + scatter: 1024 blocks x 256 threads (8 waves,
    //    16 tokens/wave -> 128 tokens/block)
    HashRouter_route_kernel<<<dim3(NUM_TOKENS / 128), dim3(256), 0, stream>>>(
        x, Wp, bp, out);
}